// VQVAE_87505663688711
// MI455X (gfx1250) — compile-verified
//
#include <hip/hip_runtime.h>
#include <math.h>

// ---------------- problem constants ----------------
#define BATCH   16
#define DMODEL  256      // d_model
#define KCB     512      // codebook size
#define HP      128      // pooled H
#define WP      128      // pooled W
#define FMS     260      // LDS A-tile row stride (pad vs 256 to break bank conflicts)
#define SCS     516      // LDS score row stride (pad vs 512)

typedef float v2f __attribute__((ext_vector_type(2)));
typedef float v8f __attribute__((ext_vector_type(8)));

// ---------------------------------------------------------------------------
// Kernel 1: c2[k] = sum_d codebook[k][d]^2      (512 x 256, trivial)
// ---------------------------------------------------------------------------
__global__ __launch_bounds__(256) void vq_c2_kernel(const float* __restrict__ cb,
                                                    float* __restrict__ c2) {
    int k = blockIdx.x * blockDim.x + threadIdx.x;
    if (k >= KCB) return;
    const float* row = cb + (size_t)k * DMODEL;
    float s = 0.f;
    #pragma unroll 8
    for (int d = 0; d < DMODEL; ++d) { float v = row[d]; s += v * v; }
    c2[k] = s;
}

// ---------------------------------------------------------------------------
// Kernel 2: relayout codebook into WMMA f32 16x16x4 B-fragment order.
// B frag for n-tile nt, k-step s: lane l holds B[k][n] with
//   n = nt*16 + (l&15),  k = 4*s + 2*(l>>4) + {0,1}
// cbB[((nt*64+s)*32 + l)*2 + j] = codebook[n*256 + k + j]
// ---------------------------------------------------------------------------
__global__ __launch_bounds__(256) void vq_relayout_kernel(const float* __restrict__ cb,
                                                          float* __restrict__ cbB) {
    int pi = blockIdx.x * blockDim.x + threadIdx.x;    // 32*64*32 = 65536
    if (pi >= 32 * 64 * 32) return;
    int l  = pi & 31;
    int s  = (pi >> 5) & 63;
    int nt = pi >> 11;
    int n  = nt * 16 + (l & 15);
    int k  = 4 * s + 2 * (l >> 4);
    v2f v = *(const v2f*)&cb[(size_t)n * DMODEL + k];
    *(v2f*)&cbB[(size_t)pi * 2] = v;
}

// ---------------------------------------------------------------------------
// Kernel 3: fully fused conv3x3 + bias + maxpool2 + BN + exact GELU + VQ.
// One workgroup (256 thr = 8 waves) per 4x4 pooled tile (16 pixels).
// ---------------------------------------------------------------------------
__global__ __launch_bounds__(256) void vq_fused_kernel(
    const float* __restrict__ x,        const float* __restrict__ conv_w,
    const float* __restrict__ conv_b,   const float* __restrict__ bn_gamma,
    const float* __restrict__ bn_beta,  const float* __restrict__ bn_mean,
    const float* __restrict__ bn_var,   const float* __restrict__ codebook,
    const float* __restrict__ cbB,      const float* __restrict__ c2,
    float* __restrict__ out_quant,      float* __restrict__ out_idx,
    float* __restrict__ out_loss)
{
    __shared__ float win[3][10][10];          // zero-padded input window
    __shared__ float fmapA[16 * FMS];         // A tile: 16 pixels x 256 ch
    __shared__ float scores[16 * SCS];        // c2 - 2*dot, 16 x 512
    __shared__ float bestv[16][16];
    __shared__ int   besti[16][16];
    __shared__ int   idxs[16];
    __shared__ float ldsLoss;

    const int t  = threadIdx.x;
    const int wg = blockIdx.x;                // 16 * 32 * 32
    const int b  = wg >> 10;
    const int by = (wg >> 5) & 31;
    const int bx = wg & 31;

    if (t == 0) ldsLoss = 0.f;

    // ---- zero-fill the padded window, then stage it with CDNA5 async
    //      global->LDS copies (ASYNCcnt path, bypasses VGPRs) ----
    float* winf = &win[0][0][0];
    for (int i = t; i < 300; i += 256) winf[i] = 0.f;
    __syncthreads();
    for (int i = t; i < 300; i += 256) {
        int ci = i / 100, rem = i % 100;
        int r = rem / 10, cc = rem % 10;
        int iy = by * 8 - 1 + r;
        int ix = bx * 8 - 1 + cc;
        if ((unsigned)iy < 256u && (unsigned)ix < 256u) {
            unsigned lds_off = (unsigned)(size_t)&win[ci][r][cc];
            unsigned long long gaddr =
                (unsigned long long)(size_t)&x[(((size_t)b * 3 + ci) * 256 + iy) * 256 + ix];
            asm volatile("global_load_async_to_lds_b32 %0, %1, off"
                         :: "v"(lds_off), "v"(gaddr) : "memory");
        }
    }
    asm volatile("s_wait_asynccnt 0x0" ::: "memory");

    // ---- per-thread channel params (c = t), weights live in 27 VGPRs ----
    const int c = t;
    float wr[27];
    #pragma unroll
    for (int j = 0; j < 27; ++j) wr[j] = conv_w[c * 27 + j];
    const float bias = conv_b[c];
    const float inv  = bn_gamma[c] * rsqrtf(bn_var[c] + 1e-5f);
    const float mu   = bn_mean[c];
    const float bet  = bn_beta[c];
    __syncthreads();

    // ---- conv + maxpool + BN + exact GELU for 16 pixels, this channel ----
    #pragma unroll
    for (int p = 0; p < 16; ++p) {
        const int py = p >> 2, px = p & 3;
        float m = -INFINITY;
        #pragma unroll
        for (int dy = 0; dy < 2; ++dy)
        #pragma unroll
        for (int dx = 0; dx < 2; ++dx) {
            const int oy = py * 2 + dy, ox = px * 2 + dx;
            float acc = bias;
            #pragma unroll
            for (int ci = 0; ci < 3; ++ci)
            #pragma unroll
            for (int ky = 0; ky < 3; ++ky)
            #pragma unroll
            for (int kx = 0; kx < 3; ++kx)
                acc = fmaf(win[ci][oy + ky][ox + kx], wr[ci * 9 + ky * 3 + kx], acc);
            m = fmaxf(m, acc);
        }
        const float v = (m - mu) * inv + bet;
        const float g = 0.5f * v * (1.0f + erff(v * 0.70710678118654752f));
        fmapA[p * FMS + c] = g;
    }
    __syncthreads();

    // ---- GEMM: 16 pixels x 512 codes, K=256, f32 WMMA 16x16x4.
    //      Each wave owns 4 independent n-tiles processed concurrently:
    //      4 accumulators -> 4-deep WMMA ILP, one shared A-frag ds_load. ----
    const int lane = t & 31;
    const int wv   = t >> 5;          // 8 waves -> n-tiles wv, wv+8, wv+16, wv+24
    const int half = lane >> 4;
    const int lr   = lane & 15;

    v8f acc0 = {0.f,0.f,0.f,0.f,0.f,0.f,0.f,0.f};
    v8f acc1 = acc0, acc2 = acc0, acc3 = acc0;
    const v2f* B0 = (const v2f*)cbB + ((size_t)wv * 64) * 32 + lane;
    #define BSTRIDE (8 * 64 * 32)     // v2f stride between the wave's n-tiles
    #pragma unroll 2
    for (int s = 0; s < 64; ++s) {
        v2f a  = *(const v2f*)&fmapA[lr * FMS + 4 * s + 2 * half];
        v2f b0 = B0[(size_t)s * 32];
        v2f b1 = B0[(size_t)s * 32 + 1 * BSTRIDE];
        v2f b2 = B0[(size_t)s * 32 + 2 * BSTRIDE];
        v2f b3 = B0[(size_t)s * 32 + 3 * BSTRIDE];
        __builtin_prefetch(&B0[(size_t)(s + 16) * 32], 0, 3);
        acc0 = __builtin_amdgcn_wmma_f32_16x16x4_f32(false, a, false, b0, (short)0, acc0, false, false);
        acc1 = __builtin_amdgcn_wmma_f32_16x16x4_f32(false, a, false, b1, (short)0, acc1, false, false);
        acc2 = __builtin_amdgcn_wmma_f32_16x16x4_f32(false, a, false, b2, (short)0, acc2, false, false);
        acc3 = __builtin_amdgcn_wmma_f32_16x16x4_f32(false, a, false, b3, (short)0, acc3, false, false);
    }
    // scores[m][n] = c2[n] - 2*dot ; D row M = j + 8*half, col n = nt*16+lr
    {
        v8f accs[4] = {acc0, acc1, acc2, acc3};
        #pragma unroll
        for (int q = 0; q < 4; ++q) {
            const int n = (wv + 8 * q) * 16 + lr;
            const float cc2 = c2[n];
            #pragma unroll
            for (int j = 0; j < 8; ++j)
                scores[(half * 8 + j) * SCS + n] = cc2 - 2.0f * accs[q][j];
        }
    }
    __syncthreads();

    // ---- two-level argmin over 512 codes per pixel ----
    {
        const int p = t >> 4, seg = t & 15;
        float bv = INFINITY; int bi = 0;
        for (int q = 0; q < 32; ++q) {
            const int n = seg * 32 + q;
            const float sv = scores[p * SCS + n];
            if (sv < bv) { bv = sv; bi = n; }
        }
        bestv[p][seg] = bv; besti[p][seg] = bi;
    }
    __syncthreads();
    if (t < 16) {
        float bv = INFINITY; int bi = 0;
        #pragma unroll
        for (int seg = 0; seg < 16; ++seg) {
            const float sv = bestv[t][seg];
            if (sv < bv) { bv = sv; bi = besti[t][seg]; }
        }
        idxs[t] = bi;
        const int ph = by * 4 + (t >> 2), pw = bx * 4 + (t & 3);
        out_idx[((size_t)b * HP + ph) * WP + pw] = (float)bi;
    }
    __syncthreads();

    // ---- gather quant rows, write NCHW output, accumulate commit loss ----
    float ls = 0.f;
    #pragma unroll 4
    for (int rep = 0; rep < 16; ++rep) {
        const int d = rep * 16 + (t >> 4);
        const int p = t & 15;
        const int ki = idxs[p];
        const float q = codebook[(size_t)ki * DMODEL + d];
        const int ph = by * 4 + (p >> 2), pw = bx * 4 + (p & 3);
        out_quant[(((size_t)b * DMODEL + d) * HP + ph) * WP + pw] = q;
        const float diff = q - fmapA[p * FMS + d];
        ls += diff * diff;
    }
    atomicAdd(&ldsLoss, ls);
    __syncthreads();
    if (t == 0) {
        // mean over B*N*D = 16*16384*256 = 67108864, COMMIT_W = 1.0
        atomicAdd(out_loss, ldsLoss * (1.0f / 67108864.0f));
    }
}

// ---------------------------------------------------------------------------
extern "C" void kernel_launch(void* const* d_in, const int* in_sizes, int n_in,
                              void* d_out, int out_size, void* d_ws, size_t ws_size,
                              hipStream_t stream) {
    const float* x        = (const float*)d_in[0];
    const float* conv_w   = (const float*)d_in[1];
    const float* conv_b   = (const float*)d_in[2];
    const float* bn_gamma = (const float*)d_in[3];
    const float* bn_beta  = (const float*)d_in[4];
    const float* bn_mean  = (const float*)d_in[5];
    const float* bn_var   = (const float*)d_in[6];
    const float* codebook = (const float*)d_in[7];

    float* out_quant = (float*)d_out;                                // 16*256*128*128
    float* out_idx   = out_quant + (size_t)BATCH * DMODEL * HP * WP; // 16*128*128
    float* out_loss  = out_idx + (size_t)BATCH * HP * WP;            // 1

    float* cbB = (float*)d_ws;                 // 32*64*32*2 floats = 512 KB
    float* c2  = cbB + 32 * 64 * 32 * 2;       // 512 floats

    hipMemsetAsync(out_loss, 0, sizeof(float), stream);

    vq_c2_kernel<<<2, 256, 0, stream>>>(codebook, c2);
    vq_relayout_kernel<<<256, 256, 0, stream>>>(codebook, cbB);
    vq_fused_kernel<<<BATCH * 32 * 32, 256, 0, stream>>>(
        x, conv_w, conv_b, bn_gamma, bn_beta, bn_mean, bn_var,
        codebook, cbB, c2, out_quant, out_idx, out_loss);
}